// Net_59906203844642
// MI455X (gfx1250) — compile-verified
//
#include <hip/hip_runtime.h>
#include <hip/hip_bf16.h>

typedef float v2f __attribute__((ext_vector_type(2)));
typedef float v8f __attribute__((ext_vector_type(8)));

#define NSEG   128
#define CCH    512
#define NPOS   4096   // 64*64
#define NBATCH 32
#define NROWS  (NBATCH * NSEG)   // 4096
#define ACCSTR 513               // padded stride: bank = (seg + c) % 64 varies per seg

// ---------------------------------------------------------------------------
// K1: per-batch segment means. One workgroup per batch, 32 waves.
// LDS: accum[128*513] + counts[128] + ids[4096]  (~273 KB of the 320 KB WGP LDS)
// ---------------------------------------------------------------------------
__global__ __launch_bounds__(1024) void k_segmeans(const float* __restrict__ x,
                                                   const long long* __restrict__ mask,
                                                   float* __restrict__ means,
                                                   float* __restrict__ score) {
  extern __shared__ float lds[];
  float* accum = lds;                         // 128*513
  float* cnts  = lds + NSEG * ACCSTR;         // 128
  int*   ids   = (int*)(cnts + NSEG);         // 4096

  const int b   = blockIdx.x;
  const int tid = threadIdx.x;

  for (int i = tid; i < NSEG * ACCSTR; i += 1024) accum[i] = 0.0f;
  if (tid < NSEG) cnts[tid] = 0.0f;
  if (tid < NSEG) score[b * NSEG + tid] = 0.0f;   // zero att-score accumulator
  __syncthreads();

  // ids (nearest-downsample gather) + histogram counts
  const long long* mb = mask + (size_t)b * 512 * 512;
  for (int p = tid; p < NPOS; p += 1024) {
    int pr = p >> 6, pc = p & 63;
    int id = (int)mb[(size_t)(pr * 8) * 512 + pc * 8];
    ids[p] = id;
    if (id > 0) atomicAdd(&cnts[id - 1], 1.0f);
  }
  __syncthreads();

  // each wave owns 16 channels; lanes sweep positions coalesced; ds_add_f32 scatter
  const int wave = tid >> 5, lane = tid & 31;
  const float* xb = x + (size_t)b * CCH * NPOS;
  for (int cc = 0; cc < 16; ++cc) {
    const int c = wave * 16 + cc;
    const float* row = xb + (size_t)c * NPOS;
    for (int p0 = 0; p0 < NPOS; p0 += 32) {
      float v = row[p0 + lane];
      int id = ids[p0 + lane];
      if (id > 0) atomicAdd(&accum[(id - 1) * ACCSTR + c], v);
    }
  }
  __syncthreads();

  // means = sums / max(counts,1)  (counts==0 -> sums==0 -> 0, matches reference)
  float* outb = means + (size_t)b * NSEG * CCH;
  for (int i = tid; i < NSEG * CCH; i += 1024) {
    int s = i >> 9, c = i & 511;
    outb[i] = accum[s * ACCSTR + c] / fmaxf(cnts[s], 1.0f);
  }
}

// ---------------------------------------------------------------------------
// K2: H = relu(means @ Wa1 + ba1); score += H @ Wa2   (fused, f32 WMMA)
// 8 waves/block, one 16x16 tile per wave. 4096x512x512 GEMM.
// ---------------------------------------------------------------------------
__global__ __launch_bounds__(256) void k_att(const float* __restrict__ means,
                                             const float* __restrict__ Wa1,
                                             const float* __restrict__ ba1,
                                             const float* __restrict__ Wa2,
                                             float* __restrict__ score) {
  const int wave = threadIdx.x >> 5;
  const int lane = threadIdx.x & 31;
  const int tile = blockIdx.x * 8 + wave;  // 0..8191
  const int m0 = (tile >> 5) * 16;
  const int n0 = (tile & 31) * 16;
  const int hi = lane >> 4, l16 = lane & 15;

  v8f acc = {};
  const float* arow = means + (size_t)(m0 + l16) * CCH;
  #pragma unroll 4
  for (int k = 0; k < CCH; k += 4) {
    const int ka = k + 2 * hi;               // lanes 0-15: K0/K1, lanes 16-31: K2/K3
    v2f a, bm;
    a.x  = arow[ka];
    a.y  = arow[ka + 1];
    bm.x = Wa1[(size_t)ka * 512 + n0 + l16];
    bm.y = Wa1[(size_t)(ka + 1) * 512 + n0 + l16];
    acc = __builtin_amdgcn_wmma_f32_16x16x4_f32(false, a, false, bm,
                                                (short)0, acc, false, false);
  }

  // fused bias + relu + (.* Wa2) + 16-lane row reduction + global atomic scatter
  const int n = n0 + l16;
  const float bn = ba1[n], w2 = Wa2[n];
  #pragma unroll
  for (int r = 0; r < 8; ++r) {
    float v = fmaxf(acc[r] + bn, 0.0f) * w2;
    #pragma unroll
    for (int off = 1; off < 16; off <<= 1) v += __shfl_xor(v, off, 32);
    if (l16 == 0) atomicAdd(&score[m0 + r + 8 * hi], v);   // rows r / r+8 per half
  }
}

// ---------------------------------------------------------------------------
// K3: cls = means @ Wh + bh  (N=19 padded to two 16-wide WMMA tiles)
// ---------------------------------------------------------------------------
__global__ __launch_bounds__(256) void k_cls(const float* __restrict__ means,
                                             const float* __restrict__ Wh,
                                             const float* __restrict__ bh,
                                             float* __restrict__ clsbuf) {
  const int wave = threadIdx.x >> 5;
  const int lane = threadIdx.x & 31;
  const int m0 = (blockIdx.x * 8 + wave) * 16;
  const int hi = lane >> 4, l16 = lane & 15;

  v8f c0 = {}, c1 = {};
  const float* arow = means + (size_t)(m0 + l16) * CCH;
  #pragma unroll 4
  for (int k = 0; k < CCH; k += 4) {
    const int ka = k + 2 * hi;
    v2f a, b0, b1;
    a.x = arow[ka];
    a.y = arow[ka + 1];
    b0.x = Wh[(size_t)ka * 19 + l16];
    b0.y = Wh[(size_t)(ka + 1) * 19 + l16];
    b1.x = (l16 < 3) ? Wh[(size_t)ka * 19 + 16 + l16] : 0.0f;
    b1.y = (l16 < 3) ? Wh[(size_t)(ka + 1) * 19 + 16 + l16] : 0.0f;
    c0 = __builtin_amdgcn_wmma_f32_16x16x4_f32(false, a, false, b0, (short)0, c0, false, false);
    c1 = __builtin_amdgcn_wmma_f32_16x16x4_f32(false, a, false, b1, (short)0, c1, false, false);
  }
  #pragma unroll
  for (int r = 0; r < 8; ++r) {
    const int row = m0 + r + 8 * hi;
    clsbuf[(size_t)row * 32 + l16] = c0[r] + bh[l16];
    if (l16 < 3) clsbuf[(size_t)row * 32 + 16 + l16] = c1[r] + bh[16 + l16];
  }
}

// ---------------------------------------------------------------------------
// K4: per-batch softmax over 128 segments + logits = sum_s att[s]*cls[s,:]
// ---------------------------------------------------------------------------
__global__ __launch_bounds__(128) void k_soft(const float* __restrict__ score,
                                              const float* __restrict__ clsbuf,
                                              const float* __restrict__ ba2,
                                              float* __restrict__ logits) {
  __shared__ float att[NSEG];
  __shared__ float red[NSEG];
  const int b = blockIdx.x, s = threadIdx.x;
  float v = score[b * NSEG + s] + ba2[0];     // constant shift (cancels in softmax)
  red[s] = v;
  __syncthreads();
  for (int off = 64; off > 0; off >>= 1) {
    if (s < off) red[s] = fmaxf(red[s], red[s + off]);
    __syncthreads();
  }
  const float m = red[0];
  __syncthreads();
  float e = expf(v - m);
  att[s] = e;
  red[s] = e;
  __syncthreads();
  for (int off = 64; off > 0; off >>= 1) {
    if (s < off) red[s] += red[s + off];
    __syncthreads();
  }
  const float inv = 1.0f / red[0];
  if (s < 19) {
    float acc = 0.0f;
    for (int t = 0; t < NSEG; ++t)
      acc += att[t] * clsbuf[(size_t)(b * NSEG + t) * 32 + s];
    logits[b * 19 + s] = acc * inv;
  }
}

// ---------------------------------------------------------------------------
// K5: mean BCE-with-logits over 32*19 elements
// ---------------------------------------------------------------------------
__global__ __launch_bounds__(256) void k_loss(const float* __restrict__ logits,
                                              const float* __restrict__ target,
                                              float* __restrict__ loss_out) {
  __shared__ float red[256];
  float acc = 0.0f;
  for (int i = threadIdx.x; i < NBATCH * 19; i += 256) {
    float l = logits[i], t = target[i];
    acc += fmaxf(l, 0.0f) - l * t + log1pf(expf(-fabsf(l)));
  }
  red[threadIdx.x] = acc;
  __syncthreads();
  for (int off = 128; off > 0; off >>= 1) {
    if (threadIdx.x < off) red[threadIdx.x] += red[threadIdx.x + off];
    __syncthreads();
  }
  if (threadIdx.x == 0) loss_out[0] = red[0] / (float)(NBATCH * 19);
}

// ---------------------------------------------------------------------------
extern "C" void kernel_launch(void* const* d_in, const int* in_sizes, int n_in,
                              void* d_out, int out_size, void* d_ws, size_t ws_size,
                              hipStream_t stream) {
  const float*     x      = (const float*)d_in[0];
  const long long* mask   = (const long long*)d_in[1];
  const float*     target = (const float*)d_in[2];
  const float*     Wa1    = (const float*)d_in[3];
  const float*     ba1    = (const float*)d_in[4];
  const float*     Wa2    = (const float*)d_in[5];
  const float*     ba2    = (const float*)d_in[6];
  const float*     Wh     = (const float*)d_in[7];
  const float*     bh     = (const float*)d_in[8];

  float* ws     = (float*)d_ws;
  float* means  = ws;                                  // 4096*512
  float* score  = ws + (size_t)NROWS * CCH;            // 4096
  float* clsbuf = score + NROWS;                       // 4096*32
  float* out    = (float*)d_out;                       // 608 logits + 1 loss

  const size_t lds1 = (size_t)(NSEG * ACCSTR + NSEG) * sizeof(float)
                    + (size_t)NPOS * sizeof(int);      // ~273 KB
  (void)hipFuncSetAttribute((const void*)k_segmeans,
                            hipFuncAttributeMaxDynamicSharedMemorySize, (int)lds1);

  k_segmeans<<<NBATCH, 1024, lds1, stream>>>(x, mask, means, score);
  k_att<<<(NROWS / 16) * (512 / 16) / 8, 256, 0, stream>>>(means, Wa1, ba1, Wa2, score);
  k_cls<<<(NROWS / 16) / 8, 256, 0, stream>>>(means, Wh, bh, clsbuf);
  k_soft<<<NBATCH, 128, 0, stream>>>(score, clsbuf, ba2, out);
  k_loss<<<1, 256, 0, stream>>>(out, target, out + NBATCH * 19);
}